// GeneticTransformer_43585328120024
// MI455X (gfx1250) — compile-verified
//
#include <hip/hip_runtime.h>
#include <math.h>

typedef __attribute__((ext_vector_type(2))) float v2f;
typedef __attribute__((ext_vector_type(8))) float v8f;

static __device__ __forceinline__ v8f wmma4(v2f a, v2f b, v8f c) {
  // D = A(16x4 f32) x B(4x16 f32) + C(16x16 f32), exact fp32 semantics
  return __builtin_amdgcn_wmma_f32_16x16x4_f32(false, a, false, b, (short)0, c,
                                               false, false);
}

#define WAVES_PER_BLOCK 8
#define LDS_PER_WAVE 1792 /* floats: 5x256 tiles + 464 logit tile */

__global__ __launch_bounds__(256) void gt_attn_kernel(
    const int* __restrict__ idx, const float* __restrict__ tok_table,
    const float* __restrict__ pos_table, const float* __restrict__ Wk,
    const float* __restrict__ Wq, const float* __restrict__ Wv,
    const float* __restrict__ lm_W, const float* __restrict__ lm_b,
    float* __restrict__ out, int n_tiles) {
  __shared__ float smem[WAVES_PER_BLOCK * LDS_PER_WAVE];
  const int lane = threadIdx.x & 31;
  const int wib = threadIdx.x >> 5;
  float* ws = smem + wib * LDS_PER_WAVE;  // wave-private: no barriers needed
  float* xs = ws;          // X tile (reused: M tile during preamble)
  float* xms = ws + 256;   // X*M tile
  float* vs = ws + 512;    // V tile
  float* es = ws + 768;    // exp(S') tile (masked, unnormalized)
  float* evs = ws + 1024;  // EV tile (unnormalized attention output)
  float* lg = ws + 1280;   // 16x29 logits tile

  const int n = lane & 15;   // N column (B/C/D frags) == M row (A frags)
  const int h2 = lane >> 4;  // wave half
  const int t = n & 7;       // token index within a batch for row n
  const bool sameblk = (((lane >> 3) & 1) == h2);  // same-batch score block
  const int col8 = lane & 7;                       // key index within block

  // ---- preamble: wave-invariant fragments (tables tiny, stay cached) ----
  const int nv = (n + 16 < 29) ? (n + 16) : 28;  // clamp: cols 29..31 unused
  v2f bv[4], bl0[4], bl1[4], posf[4], aq[4], bkf[4];
#pragma unroll
  for (int g = 0; g < 4; ++g) {
    const int kk = 4 * g + 2 * h2;  // K index of element j=0 in this slice
    bv[g].x = Wv[n * 16 + kk];      bv[g].y = Wv[n * 16 + kk + 1];
    bl0[g].x = lm_W[n * 16 + kk];   bl0[g].y = lm_W[n * 16 + kk + 1];
    bl1[g].x = lm_W[nv * 16 + kk];  bl1[g].y = lm_W[nv * 16 + kk + 1];
    posf[g].x = pos_table[t * 16 + kk];
    posf[g].y = pos_table[t * 16 + kk + 1];
    // A-frag of Wq^T and B-frag of Wk for M = Wq^T @ Wk
    aq[g].x = Wq[kk * 16 + n];       aq[g].y = Wq[(kk + 1) * 16 + n];
    bkf[g].x = Wk[kk * 16 + n];      bkf[g].y = Wk[(kk + 1) * 16 + n];
  }
  const float bias0 = lm_b[n];
  const float bias1 = lm_b[nv];

  // M = (Wq^T Wk) * 0.25 * log2(e): folds attention scale + exp2 conversion,
  // since S = Q K^T = X (Wq^T Wk) X^T.
  v8f mf = {};
#pragma unroll
  for (int g = 0; g < 4; ++g) mf = wmma4(aq[g], bkf[g], mf);
  const float mscale = 0.25f * 1.44269504088896340736f;
#pragma unroll
  for (int r = 0; r < 8; ++r) xs[(r + 8 * h2) * 16 + n] = mf[r] * mscale;
  v2f bm[4];  // B-frag of M: B[k=e][n=f] = M[e][f]
#pragma unroll
  for (int g = 0; g < 4; ++g) {
    const int kk = 4 * g + 2 * h2;
    bm[g].x = xs[kk * 16 + n];
    bm[g].y = xs[(kk + 1) * 16 + n];
  }
  v2f onesv;  onesv.x = 1.f;  onesv.y = 1.f;  // rowsum-via-WMMA operand

  const int gwave = blockIdx.x * WAVES_PER_BLOCK + wib;
  const int nwaves = gridDim.x * WAVES_PER_BLOCK;

  for (int tile = gwave; tile < n_tiles; tile += nwaves) {
    const int b0 = tile * 2;         // rows 0-7 -> batch b0, 8-15 -> b0+1
    const int id = idx[b0 * 8 + n];  // coalesced across lanes

    // ---- x = tok_table[idx] + pos, as A-fragments; stage X row-major ----
    v2f ax[4];
#pragma unroll
    for (int g = 0; g < 4; ++g) {
      const int kk = 4 * g + 2 * h2;
      const v2f tr = *(const v2f*)(tok_table + id * 16 + kk);
      ax[g].x = tr.x + posf[g].x;
      ax[g].y = tr.y + posf[g].y;
      *(v2f*)(xs + n * 16 + kk) = ax[g];
    }

    // ---- XM = X@M , V = X@Wv^T : 8 wmma ----
    v8f xmf = {}, vf = {};
#pragma unroll
    for (int g = 0; g < 4; ++g) {
      xmf = wmma4(ax[g], bm[g], xmf);
      vf = wmma4(ax[g], bv[g], vf);
    }
#pragma unroll
    for (int r = 0; r < 8; ++r) {
      const int m = r + 8 * h2;
      xms[m * 16 + n] = xmf[r];
      vs[m * 16 + n] = vf[r];
    }

    // ---- S' = XM @ X^T (pre-scaled, in log2 domain) : 4 wmma ----
    v8f sf = {};
#pragma unroll
    for (int g = 0; g < 4; ++g) {
      const int kk = 4 * g + 2 * h2;
      const v2f a = *(const v2f*)(xms + n * 16 + kk);
      const v2f b = *(const v2f*)(xs + n * 16 + kk);
      sf = wmma4(a, b, sf);
    }

    // ---- masked exp (no max-subtraction: |S*scale| << 1, overflow-free) ----
#pragma unroll
    for (int r = 0; r < 8; ++r) {
      const bool valid = sameblk && (col8 <= r);
      const float e = valid ? __builtin_amdgcn_exp2f(sf[r]) : 0.f;
      es[(r + 8 * h2) * 16 + n] = e;
    }

    // ---- EV = E@V ; R = E@ones (rowsum broadcast into same frag) : 8 wmma --
    v8f ev = {}, rs = {};
#pragma unroll
    for (int g = 0; g < 4; ++g) {
      const int kk = 4 * g + 2 * h2;
      const v2f a = *(const v2f*)(es + n * 16 + kk);
      v2f b;
      b.x = vs[kk * 16 + n];
      b.y = vs[(kk + 1) * 16 + n];
      ev = wmma4(a, b, ev);
      rs = wmma4(a, onesv, rs);
    }
    // stage UNNORMALIZED EV; normalization commutes with the logits matmul
#pragma unroll
    for (int r = 0; r < 8; ++r) evs[(r + 8 * h2) * 16 + n] = ev[r];

    // ---- logits = diag(1/R) * (EV @ lm_W^T) + bias : 8 wmma ----
    v8f l0 = {}, l1 = {};
#pragma unroll
    for (int g = 0; g < 4; ++g) {
      const int kk = 4 * g + 2 * h2;
      const v2f a = *(const v2f*)(evs + n * 16 + kk);
      l0 = wmma4(a, bl0[g], l0);
      l1 = wmma4(a, bl1[g], l1);
    }
#pragma unroll
    for (int r = 0; r < 8; ++r) {
      const int m = r + 8 * h2;
      const float rr = __builtin_amdgcn_rcpf(rs[r]);  // 1/rowsum[m]
      lg[m * 29 + n] = fmaf(l0[r], rr, bias0);
      if (n < 13) lg[m * 29 + 16 + n] = fmaf(l1[r], rr, bias1);
    }

    // ---- coalesced output: 464 floats = 116 float4, 16B-aligned base ----
    const float4* lg4 = (const float4*)lg;
    float4* dst = (float4*)(out + (size_t)tile * 464);
    for (int i = lane; i < 116; i += 32) dst[i] = lg4[i];
  }
}

extern "C" void kernel_launch(void* const* d_in, const int* in_sizes, int n_in,
                              void* d_out, int out_size, void* d_ws,
                              size_t ws_size, hipStream_t stream) {
  const int* idx = (const int*)d_in[0];
  const float* tok = (const float*)d_in[1];
  const float* pos = (const float*)d_in[2];
  const float* Wk = (const float*)d_in[3];
  const float* Wq = (const float*)d_in[4];
  const float* Wv = (const float*)d_in[5];
  const float* lmW = (const float*)d_in[6];
  const float* lmb = (const float*)d_in[7];
  float* out = (float*)d_out;

  const int B = in_sizes[0] / 8;  // BLOCK = 8
  const int n_tiles = B / 2;      // two batch rows per 16x16 tile (B is even)

  int blocks = 1024;  // 8192 waves -> 8 tiles/wave: amortizes wave preamble
  const int max_blocks = (n_tiles + WAVES_PER_BLOCK - 1) / WAVES_PER_BLOCK;
  if (blocks > max_blocks) blocks = max_blocks;
  if (blocks < 1) blocks = 1;

  gt_attn_kernel<<<blocks, 32 * WAVES_PER_BLOCK, 0, stream>>>(
      idx, tok, pos, Wk, Wq, Wv, lmW, lmb, out, n_tiles);
}